// HashGrid_33474975105224
// MI455X (gfx1250) — compile-verified
//
#include <hip/hip_runtime.h>
#include <cstdint>
#include <cstddef>

// HashGrid encode: out[n, l*2+f] = bilinear blend of 4 hashed table entries.
// T = 524288 = 2^19  -> floor-mod+abs == (bits & (T-1)) in unsigned math.
// PI1 = -1640531535 (int32) == 2654435761u ; PI2 = 805459861.

#define HG_THREADS 256

// Native clang vector types: accepted by __builtin_nontemporal_* and lower to
// b64 / b128 global loads+stores.
typedef float v2f __attribute__((ext_vector_type(2)));
typedef float v4f __attribute__((ext_vector_type(4)));

__global__ __launch_bounds__(HG_THREADS)
void hashgrid_kernel(const float* __restrict__ xy,
                     const float* __restrict__ tables,
                     float* __restrict__ out,
                     int N)
{
    // res[l] = int(16 * 1.38^l), baked at compile time (double-precision exact)
    constexpr float res[16] = {
        16.f,   22.f,   30.f,   42.f,   58.f,   80.f,  110.f,  152.f,
        210.f,  290.f,  400.f,  553.f,  763.f, 1053.f, 1453.f, 2005.f
    };

    const int n = blockIdx.x * HG_THREADS + threadIdx.x;
    if (n >= N) return;

    // One-shot streaming read of the point: non-temporal so it never competes
    // with the L2-resident tables.
    const v2f* xy2 = reinterpret_cast<const v2f*>(xy);
    v2f p = __builtin_nontemporal_load(&xy2[n]);

    const float x = p.x * 0.5f + 0.5f;   // in [0,1]
    const float y = p.y * 0.5f + 0.5f;

    v4f* outp = reinterpret_cast<v4f*>(out + (size_t)n * 32);

    const uint32_t PI1 = 2654435761u;   // == (uint32)(-1640531535)
    const uint32_t PI2 = 805459861u;
    const uint32_t TMASK = 524287u;     // 2^19 - 1

#pragma unroll
    for (int lp = 0; lp < 8; ++lp) {     // level pairs -> one b128 NT store each
        v4f r4;
#pragma unroll
        for (int s = 0; s < 2; ++s) {
            const int l = lp * 2 + s;

            const float sx = x * res[l];
            const float sy = y * res[l];
            const float flx = floorf(sx);
            const float fly = floorf(sy);
            const float fx = sx - flx;
            const float fy = sy - fly;

            const uint32_t ix = (uint32_t)(int)flx;
            const uint32_t iy = (uint32_t)(int)fly;

            // int32 wrap-around multiply == uint32 multiply (bit-identical)
            const uint32_t hx0 = ix        * PI1;
            const uint32_t hx1 = (ix + 1u) * PI1;
            const uint32_t hy0 = iy        * PI2;
            const uint32_t hy1 = (iy + 1u) * PI2;

            const uint32_t i00 = (hx0 ^ hy0) & TMASK;   // v_bitop3_b32
            const uint32_t i10 = (hx1 ^ hy0) & TMASK;
            const uint32_t i01 = (hx0 ^ hy1) & TMASK;
            const uint32_t i11 = (hx1 ^ hy1) & TMASK;

            // Level table base: compile-time constant offset after unroll.
            const v2f* tb =
                reinterpret_cast<const v2f*>(tables) + (size_t)l * 524288u;

            // Four random 8B gathers: L2-resident (64MB tables << 192MB L2).
            const v2f c00 = tb[i00];
            const v2f c10 = tb[i10];
            const v2f c01 = tb[i01];
            const v2f c11 = tb[i11];

            const float gx = 1.0f - fx;
            const float gy = 1.0f - fy;
            const float w00 = gx * gy;
            const float w10 = fx * gy;
            const float w01 = gx * fy;
            const float w11 = fx * fy;

            // Packed f32 blend (v_pk_fma_f32 / v_pk_mul_f32)
            const v2f r = w00 * c00 + w10 * c10 + w01 * c01 + w11 * c11;

            if (s == 0) { r4.x = r.x; r4.y = r.y; }
            else        { r4.z = r.x; r4.w = r.y; }
        }
        // Streaming output (128MB total): b128 non-temporal store keeps L2
        // reserved for the hash tables.
        __builtin_nontemporal_store(r4, &outp[lp]);
    }
}

extern "C" void kernel_launch(void* const* d_in, const int* in_sizes, int n_in,
                              void* d_out, int out_size, void* d_ws, size_t ws_size,
                              hipStream_t stream)
{
    const float* xy     = (const float*)d_in[0];   // (N, 2) f32
    const float* tables = (const float*)d_in[1];   // (16, 524288, 2) f32
    float* out          = (float*)d_out;           // (N, 32) f32

    const int N = in_sizes[0] / 2;
    const int blocks = (N + HG_THREADS - 1) / HG_THREADS;

    hashgrid_kernel<<<blocks, HG_THREADS, 0, stream>>>(xy, tables, out, N);
}